// CrossAttentionModel_33569464386136
// MI455X (gfx1250) — compile-verified
//
#include <hip/hip_runtime.h>
#include <hip/hip_bf16.h>

// Problem constants (match reference)
#define BATCH 16
#define NMEM  8192
#define EMB   1024
#define TOPK  8
#define NCHUNK 64                  // chunks per batch in the streaming pass
#define ROWS_PER_CHUNK (NMEM / NCHUNK)   // 128
#define INV_SCALE (1.0f / 32.0f)   // 1/sqrt(1024)

typedef __attribute__((ext_vector_type(2))) float v2f;
typedef __attribute__((ext_vector_type(4))) float f32x4;
typedef __attribute__((ext_vector_type(8))) float v8f;

__device__ __forceinline__ float dot4(f32x4 a, f32x4 b) {
    f32x4 p = a * b;
    return p.x + p.y + p.z + p.w;
}

// ---------------------------------------------------------------------------
// Small projection GEMM via WMMA: Out[16,1024] = A[16,1024] x W' (+bias)*scale
// TRANSW=1: B[k][n] = W[n*E + k]   (i.e. A @ W^T)  -> b64 vector loads
// TRANSW=0: B[k][n] = W[k*E + n]   (i.e. A @ W)    -> two strided b32 loads
// One wave per 16-wide N tile; K swept with v_wmma_f32_16x16x4_f32.
// TRANSW is a template parameter so the K-loop is branch-free.
// ---------------------------------------------------------------------------
template <int TRANSW>
__global__ __launch_bounds__(32) void proj16_wmma(
    const float* __restrict__ A, const float* __restrict__ W,
    const float* __restrict__ bias, float* __restrict__ Out,
    float outScale)
{
    const int E = EMB;
    int ntile = blockIdx.x;            // 0..63
    int l  = threadIdx.x;              // 0..31
    int mn = l & 15;                   // A row (M) / B,D column (N)
    int k0 = (l >> 4) * 2;             // lane K offset within 4-step: 0 or 2
    int nglob = ntile * 16 + mn;

    const float* Arow = A + mn * E + k0;            // consecutive pair per step
    const float* Wrow = TRANSW ? (W + nglob * E + k0)   // consecutive pair
                               : (W + k0 * E + nglob);  // stride-E pair

    v8f c = {};
    for (int k = 0; k < E; k += 16) {
#pragma unroll
        for (int kk = 0; kk < 16; kk += 4) {
            v2f a = *(const v2f*)(Arow + k + kk);
            v2f b;
            if (TRANSW) {
                b = *(const v2f*)(Wrow + k + kk);
            } else {
                b.x = Wrow[(k + kk) * E];
                b.y = Wrow[(k + kk) * E + E];
            }
            c = __builtin_amdgcn_wmma_f32_16x16x4_f32(
                    false, a, false, b, (short)0, c, false, false);
        }
    }
    float bb = bias ? bias[nglob] : 0.0f;
    int mbase = (l >> 4) * 8;          // D: lanes 0-15 -> M 0..7, lanes 16-31 -> M 8..15
#pragma unroll
    for (int vg = 0; vg < 8; ++vg)
        Out[(mbase + vg) * E + nglob] = c[vg] * outScale + bb;
}

// c_b = (q_b . bk) / 32
__global__ __launch_bounds__(256) void dot_qbk(
    const float* __restrict__ q, const float* __restrict__ bk, float* __restrict__ qbk)
{
    __shared__ float red[256];
    int b = blockIdx.x, t = threadIdx.x;
    float s = 0.f;
    for (int f = t; f < EMB; f += 256) s += q[b * EMB + f] * bk[f];
    red[t] = s; __syncthreads();
    for (int off = 128; off > 0; off >>= 1) {
        if (t < off) red[t] += red[t + off];
        __syncthreads();
    }
    if (t == 0) qbk[b] = red[0] * INV_SCALE;
}

// ---------------------------------------------------------------------------
// Fused streaming pass: single read of memory_values (512 MB).
// Per (batch, chunk) WG: scores_n = qhat_b . mem_bn + c_b  (written out raw),
// plus flash-style online softmax accumulator acc_f = sum exp(s-m)*mem[n,f]
// kept in 32 VGPRs/lane. Waves combine via LDS float atomics.
// ---------------------------------------------------------------------------
__global__ __launch_bounds__(256) void attn_stream(
    const float* __restrict__ mem, const float* __restrict__ qhat,
    const float* __restrict__ qbk, float* __restrict__ scoresOut,
    float* __restrict__ mC, float* __restrict__ sC, float* __restrict__ accC)
{
    __shared__ float mW[8], sW[8];
    __shared__ float accBuf[EMB];

    int b = blockIdx.x >> 6;           // /NCHUNK
    int c = blockIdx.x & (NCHUNK - 1);
    int wave = threadIdx.x >> 5;       // 0..7
    int lane = threadIdx.x & 31;

    // Each lane owns features f = lane*4 + 128*i (+0..3), i = 0..7 (1024 total/wave)
    f32x4 qh[8];
    const f32x4* q4 = (const f32x4*)(qhat + b * EMB);
#pragma unroll
    for (int i = 0; i < 8; ++i) qh[i] = q4[lane + 32 * i];
    float cb = qbk[b];

    f32x4 zero = {0.f, 0.f, 0.f, 0.f};
    f32x4 acc[8];
#pragma unroll
    for (int i = 0; i < 8; ++i) acc[i] = zero;
    float m = -3.4e38f, s = 0.f;

    const float* base = mem + ((size_t)b * NMEM + (size_t)c * ROWS_PER_CHUNK) * EMB;
    for (int r = 0; r < ROWS_PER_CHUNK / 8; ++r) {
        int n = wave + 8 * r;
        const f32x4* row4 = (const f32x4*)(base + (size_t)n * EMB);
        f32x4 x[8];
#pragma unroll
        for (int i = 0; i < 8; ++i)
            x[i] = __builtin_nontemporal_load(row4 + lane + 32 * i);
        float d = 0.f;
#pragma unroll
        for (int i = 0; i < 8; ++i) d += dot4(x[i], qh[i]);
        // wave32 butterfly: broadcast full sum to all lanes
        for (int off = 16; off >= 1; off >>= 1) d += __shfl_xor(d, off, 32);
        float score = d + cb;
        if (lane == 0)
            scoresOut[b * NMEM + c * ROWS_PER_CHUNK + n] = score;
        // online softmax update (uniform across lanes)
        float mNew = fmaxf(m, score);
        float so = __expf(m - mNew);      // 0 on first iter (m = -inf-ish)
        float w  = __expf(score - mNew);
        s = s * so + w;
#pragma unroll
        for (int i = 0; i < 8; ++i) acc[i] = acc[i] * so + x[i] * w;
        m = mNew;
    }

    // combine 8 waves
    for (int t = threadIdx.x; t < EMB; t += 256) accBuf[t] = 0.f;
    if (lane == 0) { mW[wave] = m; sW[wave] = s; }
    __syncthreads();
    float Mc = -3.4e38f;
#pragma unroll
    for (int j = 0; j < 8; ++j) Mc = fmaxf(Mc, mW[j]);
    float Sc = 0.f;
#pragma unroll
    for (int j = 0; j < 8; ++j) Sc += sW[j] * __expf(mW[j] - Mc);
    float rw = __expf(m - Mc);
#pragma unroll
    for (int i = 0; i < 8; ++i) {
        int f = lane * 4 + 128 * i;
        f32x4 a = acc[i] * rw;
        atomicAdd(&accBuf[f + 0], a.x);
        atomicAdd(&accBuf[f + 1], a.y);
        atomicAdd(&accBuf[f + 2], a.z);
        atomicAdd(&accBuf[f + 3], a.w);
    }
    __syncthreads();
    if (threadIdx.x == 0) { mC[b * NCHUNK + c] = Mc; sC[b * NCHUNK + c] = Sc; }
    float* accOut = accC + (size_t)(b * NCHUNK + c) * EMB;
    for (int t = threadIdx.x; t < EMB; t += 256) accOut[t] = accBuf[t];
}

// Combine chunks: global M,S per batch; ptil_f = (1/S) sum_c acc_c[f]*exp(m_c-M)
__global__ __launch_bounds__(256) void finalize_chunks(
    const float* __restrict__ mC, const float* __restrict__ sC,
    const float* __restrict__ accC, float* __restrict__ Mg,
    float* __restrict__ Sg, float* __restrict__ ptil)
{
    int b = blockIdx.x, t = threadIdx.x;
    float M = -3.4e38f;
    for (int j = 0; j < NCHUNK; ++j) M = fmaxf(M, mC[b * NCHUNK + j]);
    float S = 0.f;
    for (int j = 0; j < NCHUNK; ++j) S += sC[b * NCHUNK + j] * __expf(mC[b * NCHUNK + j] - M);
    if (t == 0) { Mg[b] = M; Sg[b] = S; }
    int f = t * 4;
    f32x4 sum = {0.f, 0.f, 0.f, 0.f};
    for (int j = 0; j < NCHUNK; ++j) {
        float w = __expf(mC[b * NCHUNK + j] - M);
        f32x4 a = *(const f32x4*)(accC + (size_t)(b * NCHUNK + j) * EMB + f);
        sum += a * w;
    }
    f32x4 r = sum * (1.f / S);
    *(f32x4*)(ptil + b * EMB + f) = r;
}

// In-place: probs = exp(score - M_b) / S_b
__global__ __launch_bounds__(256) void probs_fin(
    float* __restrict__ sp, const float* __restrict__ Mg, const float* __restrict__ Sg)
{
    int i = blockIdx.x * 256 + threadIdx.x;     // < BATCH*NMEM
    int b = i >> 13;
    sp[i] = __expf(sp[i] - Mg[b]) / Sg[b];
}

// 8-round argmax (ties -> lowest index, matching jax top_k)
__global__ __launch_bounds__(256) void topk8(
    const float* __restrict__ probs, int* __restrict__ outIdx)
{
    __shared__ float row[NMEM];
    __shared__ float rv[256];
    __shared__ int   ri[256];
    int b = blockIdx.x, t = threadIdx.x;
    for (int j = t; j < NMEM; j += 256) row[j] = probs[b * NMEM + j];
    __syncthreads();
    for (int k = 0; k < TOPK; ++k) {
        float bv = -1.f; int bi = NMEM;
        for (int j = t; j < NMEM; j += 256) {
            float v = row[j];
            if (v > bv || (v == bv && j < bi)) { bv = v; bi = j; }
        }
        rv[t] = bv; ri[t] = bi; __syncthreads();
        for (int off = 128; off > 0; off >>= 1) {
            if (t < off) {
                float ov = rv[t + off]; int oi = ri[t + off];
                if (ov > rv[t] || (ov == rv[t] && oi < ri[t])) { rv[t] = ov; ri[t] = oi; }
            }
            __syncthreads();
        }
        if (t == 0) { outIdx[b * TOPK + k] = ri[0]; row[ri[0]] = -1.f; }
        __syncthreads();
    }
}

// top_memories[b,j,:] = memory_values[b, idx[b,j], :]
__global__ __launch_bounds__(256) void gather_top(
    const float* __restrict__ mem, const int* __restrict__ idx, float* __restrict__ outTM)
{
    int b = blockIdx.x >> 3, j = blockIdx.x & 7;
    int n = idx[b * TOPK + j];
    const f32x4* src = (const f32x4*)(mem + ((size_t)b * NMEM + (size_t)n) * EMB);
    f32x4*       dst = (f32x4*)(outTM + ((size_t)b * TOPK + j) * EMB);
    dst[threadIdx.x] = src[threadIdx.x];       // 256 * 16B = 4KB
}

extern "C" void kernel_launch(void* const* d_in, const int* in_sizes, int n_in,
                              void* d_out, int out_size, void* d_ws, size_t ws_size,
                              hipStream_t stream) {
    (void)in_sizes; (void)n_in; (void)out_size; (void)ws_size;
    const float* query = (const float*)d_in[0];
    const float* mem   = (const float*)d_in[1];
    const float* Wq    = (const float*)d_in[2];
    const float* bq    = (const float*)d_in[3];
    const float* Wk    = (const float*)d_in[4];
    const float* bk    = (const float*)d_in[5];
    const float* Wv    = (const float*)d_in[6];
    const float* bv    = (const float*)d_in[7];
    // d_in[8] = top_k (fixed at 8 per reference)

    float* out   = (float*)d_out;
    float* outO  = out;                                  // output      [16,1024]
    float* outP  = out + BATCH * EMB;                    // probs       [16,8192]
    float* outTM = outP + BATCH * NMEM;                  // top_memories[16,8,1024]
    int*   outTI = (int*)(outTM + BATCH * TOPK * EMB);   // top_indices [16,8] (int32)

    float* ws   = (float*)d_ws;
    float* q    = ws;                        // [16,1024]
    float* qhat = q    + BATCH * EMB;        // [16,1024]  (q @ Wk)/32
    float* qbk  = qhat + BATCH * EMB;        // [16]       (q . bk)/32
    float* mC   = qbk  + BATCH;              // [16,64]
    float* sC   = mC   + BATCH * NCHUNK;     // [16,64]
    float* accC = sC   + BATCH * NCHUNK;     // [16,64,1024] = 4 MB
    float* Mg   = accC + (size_t)BATCH * NCHUNK * EMB;   // [16]
    float* Sg   = Mg   + BATCH;              // [16]
    float* ptil = Sg   + BATCH;              // [16,1024]  probs @ mem

    // 1) q = query @ Wq^T + bq           (WMMA)
    proj16_wmma<1><<<EMB / 16, 32, 0, stream>>>(query, Wq, bq, q, 1.0f);
    // 2) qhat = (q @ Wk) / sqrt(E)       (WMMA)
    proj16_wmma<0><<<EMB / 16, 32, 0, stream>>>(q, Wk, nullptr, qhat, INV_SCALE);
    // 3) qbk_b = (q_b . bk) / sqrt(E)
    dot_qbk<<<BATCH, 256, 0, stream>>>(q, bk, qbk);
    // 4) single 512MB stream over memory_values: raw scores + online-softmax partials
    attn_stream<<<BATCH * NCHUNK, 256, 0, stream>>>(mem, qhat, qbk, outP, mC, sC, accC);
    // 5) combine chunks -> M, S, ptil = probs @ mem
    finalize_chunks<<<BATCH, 256, 0, stream>>>(mC, sC, accC, Mg, Sg, ptil);
    // 6) probs = exp(score - M)/S, in place in d_out
    probs_fin<<<(BATCH * NMEM) / 256, 256, 0, stream>>>(outP, Mg, Sg);
    // 7) output = ptil @ Wv^T + bv       (WMMA)
    proj16_wmma<1><<<EMB / 16, 32, 0, stream>>>(ptil, Wv, bv, outO, 1.0f);
    // 8) top-8 indices per batch
    topk8<<<BATCH, 256, 0, stream>>>(outP, outTI);
    // 9) gather top memories
    gather_top<<<BATCH * TOPK, 256, 0, stream>>>(mem, outTI, outTM);
}